// LinAttLayer_49065706389742
// MI455X (gfx1250) — compile-verified
//
#include <hip/hip_runtime.h>
#include <hip/hip_bf16.h>

#define B_ 8
#define D_ 128
#define T_ 32768
#define C_ 64
#define TT 64

#define XS_STRIDE (TT + 16)   // 80 floats: row-pair bank offset 2*80%64=32 -> conflict-free halves
#define VS_STRIDE (TT + 20)   // 84 floats: m vs m+8 store offset 8*84%64=32 -> conflict-free
#define QT_STRIDE (C_ + 1)    // 65 floats: transposed tiles, lane-consecutive banks

#define ITERS_A 16
#define WGX_A (T_ / (TT * ITERS_A))   // 32
#define ITERS_B 16
#define WGX_B (T_ / (TT * ITERS_B))   // 32

typedef float v2f __attribute__((ext_vector_type(2)));
typedef float v8f __attribute__((ext_vector_type(8)));

__device__ __forceinline__ v8f wmma_f32_k4(v2f a, v2f b, v8f c) {
  // D(16x16,f32) = A(16x4,f32) * B(4x16,f32) + C
  return __builtin_amdgcn_wmma_f32_16x16x4_f32(false, a, false, b, (short)0, c,
                                               false, false);
}

__device__ __forceinline__ v8f zero_v8f() {
  v8f z = {0.f, 0.f, 0.f, 0.f, 0.f, 0.f, 0.f, 0.f};
  return z;
}

// ---- Async Global->LDS copy (CDNA5 Tensor/async path, tracked by ASYNCcnt) ----
// Each lane copies 16 bytes from the global address g to the LDS address lds32. EXEC-driven.
__device__ __forceinline__ void async_copy_b128(const float* g, const float* l) {
  unsigned lds32 = (unsigned)(uintptr_t)l;   // generic shared ptr: low 32 bits = LDS addr
  asm volatile("global_load_async_to_lds_b128 %0, %1, off"
               :
               : "v"(lds32), "v"(g)
               : "memory");
}
__device__ __forceinline__ void wait_asynccnt_le8() {
  asm volatile("s_wait_asynccnt 0x8" ::: "memory");
}
__device__ __forceinline__ void wait_asynccnt_0() {
  asm volatile("s_wait_asynccnt 0x0" ::: "memory");
}

// Issue one X tile (128 rows x 64 cols f32) as 8 async b128 copies per lane.
__device__ __forceinline__ void issue_x_tile(const float* __restrict__ gbase,
                                             float (*Xbuf)[XS_STRIDE], int tid) {
#pragma unroll
  for (int i = 0; i < 8; ++i) {
    int idx = tid + i * 256;
    int d = idx >> 4;
    int c4 = (idx & 15) * 4;
    async_copy_b128(gbase + (size_t)d * T_ + c4, &Xbuf[d][c4]);
  }
}

// -------------------- Pass A: S[b] += Q @ Kp^T over a T slab --------------------
__global__ __launch_bounds__(256) void linatt_passA(
    const float* __restrict__ x1,
    const float* __restrict__ wq, const float* __restrict__ bq,
    const float* __restrict__ wk, const float* __restrict__ bk,
    const float* __restrict__ we, const float* __restrict__ be,
    float* __restrict__ gS) {
  __shared__ float Xs[2][D_][XS_STRIDE]; // double-buffered X tile [128 x 64]
  __shared__ float Ks[C_][VS_STRIDE];    // K = Wk@X + bk (row major)
  __shared__ float Qt[TT][QT_STRIDE];    // Q transposed: Qt[t][m]
  __shared__ float Kpt[TT][QT_STRIDE];   // Kp transposed: Kpt[t][n]

  const int b = blockIdx.y;
  const int tid = threadIdx.x;
  const int lane = tid & 31;
  const int wave = tid >> 5;
  const int frow = lane & 15;          // M (A-frag) / N (B,C,D frags)
  const int fk = (lane >> 4) * 2;      // K base for A/B frags
  const int half8 = (lane >> 4) * 8;   // M offset for C/D frags
  const int mt = wave & 3;

  const int t00 = blockIdx.x * (TT * ITERS_A);
  const float* xb = x1 + (size_t)(b * D_) * T_;

  // Prologue: start async DMA of tile 0 while we pull weights into registers.
  issue_x_tile(xb + t00, Xs[0], tid);

  // Preload weight A-fragments into registers (each wave owns one M tile).
  const float* Wmat = (wave < 4) ? wq : wk;
  const float* bvec = (wave < 4) ? bq : bk;
  v2f wfrag[32];
#pragma unroll
  for (int kk = 0; kk < 32; ++kk) {
    const float* p = Wmat + (mt * 16 + frow) * D_ + kk * 4 + fk;
    wfrag[kk].x = p[0];
    wfrag[kk].y = p[1];
  }
  v2f wefrag[16];
#pragma unroll
  for (int kk = 0; kk < 16; ++kk) {
    const float* p = we + (mt * 16 + frow) * C_ + kk * 4 + fk;
    wefrag[kk].x = p[0];
    wefrag[kk].y = p[1];
  }
  float qkb[8], beb[8];
#pragma unroll
  for (int r = 0; r < 8; ++r) {
    qkb[r] = bvec[mt * 16 + r + half8];
    beb[r] = be[mt * 16 + r + half8];
  }

  // S accumulators: wave handles 16x16 tiles (2w) and (2w+1) of the 4x4 grid.
  v8f sacc0 = zero_v8f();
  v8f sacc1 = zero_v8f();

  for (int it = 0; it < ITERS_A; ++it) {
    const int cur = it & 1;
    const int nxt = cur ^ 1;

    // Prefetch next tile into the other buffer, then wait for the current one.
    if (it + 1 < ITERS_A) {
      issue_x_tile(xb + t00 + (it + 1) * TT, Xs[nxt], tid);
      wait_asynccnt_le8();   // in-order completion: current tile's 8 are done
    } else {
      wait_asynccnt_0();
    }
    __syncthreads();         // all waves' current-tile DMA visible

    // GEMM 1/2: waves 0-3 -> Q = Wq@X + bq (store transposed),
    //           waves 4-7 -> K = Wk@X + bk (store row-major).
#pragma unroll
    for (int tt = 0; tt < 4; ++tt) {
      v8f acc = zero_v8f();
#pragma unroll
      for (int kk = 0; kk < 32; ++kk) {
        v2f bfr;
        bfr.x = Xs[cur][kk * 4 + fk][tt * 16 + frow];
        bfr.y = Xs[cur][kk * 4 + fk + 1][tt * 16 + frow];
        acc = wmma_f32_k4(wfrag[kk], bfr, acc);
      }
      const int tcol = tt * 16 + frow;
      if (wave < 4) {
#pragma unroll
        for (int r = 0; r < 8; ++r)
          Qt[tcol][mt * 16 + r + half8] = acc[r] + qkb[r];
      } else {
#pragma unroll
        for (int r = 0; r < 8; ++r)
          Ks[mt * 16 + r + half8][tcol] = acc[r] + qkb[r];
      }
    }
    __syncthreads();

    // GEMM 3: Kp = We@K + be, stored transposed. Wave -> mtile mt, 2 t-tiles.
#pragma unroll
    for (int i = 0; i < 2; ++i) {
      const int tt = (wave >> 2) + i * 2;
      v8f acc = zero_v8f();
#pragma unroll
      for (int kk = 0; kk < 16; ++kk) {
        v2f bfr;
        bfr.x = Ks[kk * 4 + fk][tt * 16 + frow];
        bfr.y = Ks[kk * 4 + fk + 1][tt * 16 + frow];
        acc = wmma_f32_k4(wefrag[kk], bfr, acc);
      }
      const int tcol = tt * 16 + frow;
#pragma unroll
      for (int r = 0; r < 8; ++r)
        Kpt[tcol][mt * 16 + r + half8] = acc[r] + beb[r];
    }
    __syncthreads();

    // GEMM 4: S += Q @ Kp^T (contraction over t within the tile).
#pragma unroll
    for (int i = 0; i < 2; ++i) {
      const int tile = wave * 2 + i;
      const int mts = tile >> 2;
      const int nts = tile & 3;
      v8f acc = (i == 0) ? sacc0 : sacc1;
#pragma unroll
      for (int kk = 0; kk < 16; ++kk) {
        v2f a, bfr;
        a.x = Qt[kk * 4 + fk][mts * 16 + frow];
        a.y = Qt[kk * 4 + fk + 1][mts * 16 + frow];
        bfr.x = Kpt[kk * 4 + fk][nts * 16 + frow];
        bfr.y = Kpt[kk * 4 + fk + 1][nts * 16 + frow];
        acc = wmma_f32_k4(a, bfr, acc);
      }
      if (i == 0) sacc0 = acc; else sacc1 = acc;
    }
    __syncthreads();
  }

  // Accumulate partial S to global scratch.
  float* Sb = gS + b * (C_ * C_);
#pragma unroll
  for (int i = 0; i < 2; ++i) {
    const int tile = wave * 2 + i;
    const int mts = tile >> 2;
    const int nts = tile & 3;
    v8f acc = (i == 0) ? sacc0 : sacc1;
#pragma unroll
    for (int r = 0; r < 8; ++r) {
      const int m = mts * 16 + r + half8;
      const int n = nts * 16 + frow;
      atomicAdd(&Sb[m * C_ + n], acc[r]);
    }
  }
}

// -------------------- Softmax over rows of S (scale 1/sqrt(64)) --------------------
__global__ __launch_bounds__(32) void linatt_softmax(const float* __restrict__ gS,
                                                     float* __restrict__ gP) {
  const int row = blockIdx.x;   // b*64 + m, 512 rows
  const int lane = threadIdx.x; // 0..31
  const float* s = gS + (size_t)row * C_;
  float v0 = s[lane] * 0.125f;
  float v1 = s[lane + 32] * 0.125f;
  float mx = fmaxf(v0, v1);
#pragma unroll
  for (int off = 16; off >= 1; off >>= 1)
    mx = fmaxf(mx, __shfl_xor(mx, off, 32));
  float e0 = __expf(v0 - mx);
  float e1 = __expf(v1 - mx);
  float sum = e0 + e1;
#pragma unroll
  for (int off = 16; off >= 1; off >>= 1)
    sum += __shfl_xor(sum, off, 32);
  const float inv = 1.0f / sum;
  float* p = gP + (size_t)row * C_;
  p[lane] = e0 * inv;
  p[lane + 32] = e1 * inv;
}

// -------------------- Pass B: out = mask * (Wo @ relu(P @ (Wf@(Wv@X+bv)+bf)) + bo) ----
__global__ __launch_bounds__(256) void linatt_passB(
    const float* __restrict__ x1, const float* __restrict__ mask,
    const float* __restrict__ wv, const float* __restrict__ bv,
    const float* __restrict__ wfm, const float* __restrict__ bfv,
    const float* __restrict__ wo, const float* __restrict__ bo,
    const float* __restrict__ gP, float* __restrict__ out) {
  __shared__ float Xs[2][D_][XS_STRIDE]; // double-buffered X tile; cur also used as out staging
  __shared__ float Vs[C_][VS_STRIDE];
  __shared__ float Vps[C_][VS_STRIDE];
  __shared__ float O1s[C_][VS_STRIDE];
  __shared__ float Ms[TT];

  const int b = blockIdx.y;
  const int tid = threadIdx.x;
  const int lane = tid & 31;
  const int wave = tid >> 5;
  const int frow = lane & 15;
  const int fk = (lane >> 4) * 2;
  const int half8 = (lane >> 4) * 8;
  const int mt = wave & 3;
  const int tb = (wave >> 2) * 2;

  const int t00 = blockIdx.x * (TT * ITERS_B);
  const float* xb = x1 + (size_t)(b * D_) * T_;

  // Prologue DMA of tile 0, overlapped with weight preloads.
  issue_x_tile(xb + t00, Xs[0], tid);

  v2f wvfrag[32];
#pragma unroll
  for (int kk = 0; kk < 32; ++kk) {
    const float* p = wv + (mt * 16 + frow) * D_ + kk * 4 + fk;
    wvfrag[kk].x = p[0];
    wvfrag[kk].y = p[1];
  }
  v2f wffrag[16];
#pragma unroll
  for (int kk = 0; kk < 16; ++kk) {
    const float* p = wfm + (mt * 16 + frow) * C_ + kk * 4 + fk;
    wffrag[kk].x = p[0];
    wffrag[kk].y = p[1];
  }
  v2f pfrag[16];
#pragma unroll
  for (int kk = 0; kk < 16; ++kk) {
    const float* p = gP + (size_t)b * (C_ * C_) + (mt * 16 + frow) * C_ + kk * 4 + fk;
    pfrag[kk].x = p[0];
    pfrag[kk].y = p[1];
  }
  v2f wofrag[16];
#pragma unroll
  for (int kk = 0; kk < 16; ++kk) {
    const float* p = wo + (wave * 16 + frow) * C_ + kk * 4 + fk;
    wofrag[kk].x = p[0];
    wofrag[kk].y = p[1];
  }
  float bvv[8], bff[8], bob[8];
#pragma unroll
  for (int r = 0; r < 8; ++r) {
    bvv[r] = bv[mt * 16 + r + half8];
    bff[r] = bfv[mt * 16 + r + half8];
    bob[r] = bo[wave * 16 + r + half8];
  }

  for (int it = 0; it < ITERS_B; ++it) {
    const int cur = it & 1;
    const int nxt = cur ^ 1;
    const int t0 = t00 + it * TT;

    if (it + 1 < ITERS_B) {
      issue_x_tile(xb + t0 + TT, Xs[nxt], tid);
      wait_asynccnt_le8();
    } else {
      wait_asynccnt_0();
    }
    if (tid < TT) Ms[tid] = mask[(size_t)b * T_ + t0 + tid];
    __syncthreads();

    // V = Wv @ X + bv
#pragma unroll
    for (int i = 0; i < 2; ++i) {
      const int tt = tb + i;
      v8f acc = zero_v8f();
#pragma unroll
      for (int kk = 0; kk < 32; ++kk) {
        v2f bfr;
        bfr.x = Xs[cur][kk * 4 + fk][tt * 16 + frow];
        bfr.y = Xs[cur][kk * 4 + fk + 1][tt * 16 + frow];
        acc = wmma_f32_k4(wvfrag[kk], bfr, acc);
      }
      const int tcol = tt * 16 + frow;
#pragma unroll
      for (int r = 0; r < 8; ++r)
        Vs[mt * 16 + r + half8][tcol] = acc[r] + bvv[r];
    }
    __syncthreads();

    // Vp = Wf @ V + bf
#pragma unroll
    for (int i = 0; i < 2; ++i) {
      const int tt = tb + i;
      v8f acc = zero_v8f();
#pragma unroll
      for (int kk = 0; kk < 16; ++kk) {
        v2f bfr;
        bfr.x = Vs[kk * 4 + fk][tt * 16 + frow];
        bfr.y = Vs[kk * 4 + fk + 1][tt * 16 + frow];
        acc = wmma_f32_k4(wffrag[kk], bfr, acc);
      }
      const int tcol = tt * 16 + frow;
#pragma unroll
      for (int r = 0; r < 8; ++r)
        Vps[mt * 16 + r + half8][tcol] = acc[r] + bff[r];
    }
    __syncthreads();

    // O1 = relu(P @ Vp)
#pragma unroll
    for (int i = 0; i < 2; ++i) {
      const int tt = tb + i;
      v8f acc = zero_v8f();
#pragma unroll
      for (int kk = 0; kk < 16; ++kk) {
        v2f bfr;
        bfr.x = Vps[kk * 4 + fk][tt * 16 + frow];
        bfr.y = Vps[kk * 4 + fk + 1][tt * 16 + frow];
        acc = wmma_f32_k4(pfrag[kk], bfr, acc);
      }
      const int tcol = tt * 16 + frow;
#pragma unroll
      for (int r = 0; r < 8; ++r)
        O1s[mt * 16 + r + half8][tcol] = fmaxf(acc[r], 0.0f);
    }
    __syncthreads();

    // Out = Wo @ O1 + bo, mask folded in -> stage into Xs[cur] (X tile is dead).
#pragma unroll
    for (int tt = 0; tt < 4; ++tt) {
      v8f acc = zero_v8f();
#pragma unroll
      for (int kk = 0; kk < 16; ++kk) {
        v2f bfr;
        bfr.x = O1s[kk * 4 + fk][tt * 16 + frow];
        bfr.y = O1s[kk * 4 + fk + 1][tt * 16 + frow];
        acc = wmma_f32_k4(wofrag[kk], bfr, acc);
      }
      const int tcol = tt * 16 + frow;
      const float mval = Ms[tcol];
#pragma unroll
      for (int r = 0; r < 8; ++r)
        Xs[cur][wave * 16 + r + half8][tcol] = (acc[r] + bob[r]) * mval;
    }
    __syncthreads();

    // 128B-coalesced store to global.
#pragma unroll
    for (int i = 0; i < 8; ++i) {
      int idx = tid + i * 256;
      int d = idx >> 4;
      int c4 = (idx & 15) * 4;
      float4 v = *(const float4*)(&Xs[cur][d][c4]);
      *(float4*)(out + (size_t)(b * D_ + d) * T_ + t0 + c4) = v;
    }
    __syncthreads();
  }
}

// -------------------- Host launch --------------------
extern "C" void kernel_launch(void* const* d_in, const int* in_sizes, int n_in,
                              void* d_out, int out_size, void* d_ws, size_t ws_size,
                              hipStream_t stream) {
  (void)in_sizes; (void)n_in; (void)out_size; (void)ws_size;

  const float* x1   = (const float*)d_in[0];
  // d_in[1] = x2 (unused in encoder stage)
  const float* mask = (const float*)d_in[2];
  const float* wq   = (const float*)d_in[3];
  const float* bq   = (const float*)d_in[4];
  const float* wk   = (const float*)d_in[5];
  const float* bk   = (const float*)d_in[6];
  const float* wv   = (const float*)d_in[7];
  const float* bv   = (const float*)d_in[8];
  const float* we   = (const float*)d_in[9];
  const float* be   = (const float*)d_in[10];
  const float* wf   = (const float*)d_in[11];
  const float* bf   = (const float*)d_in[12];
  const float* wo   = (const float*)d_in[13];
  const float* bo   = (const float*)d_in[14];
  float* outp = (float*)d_out;

  float* gS = (float*)d_ws;                 // 8*64*64 f32
  float* gP = gS + B_ * C_ * C_;            // 8*64*64 f32

  hipMemsetAsync(gS, 0, (size_t)B_ * C_ * C_ * sizeof(float), stream);
  linatt_passA<<<dim3(WGX_A, B_), 256, 0, stream>>>(x1, wq, bq, wk, bk, we, be, gS);
  linatt_softmax<<<B_ * C_, 32, 0, stream>>>(gS, gP);
  linatt_passB<<<dim3(WGX_B, B_), 256, 0, stream>>>(x1, mask, wv, bv, wf, bf, wo, bo,
                                                    gP, outp);
}